// Recommender_87239375716570
// MI455X (gfx1250) — compile-verified
//
#include <hip/hip_runtime.h>
#include <math.h>

#define N_USERS    100000
#define N_ITEMS    50000
#define N_ENTITIES 200000
#define N_RELATIONS 32
#define N_EDGES    2000000
#define NNZ        1000000
#define DIM        64
#define N_HOPS     2

typedef float v2f __attribute__((ext_vector_type(2)));
typedef float v8f __attribute__((ext_vector_type(8)));

// ---------------------------------------------------------------- zero scratch
__global__ void zero_kernel(float* __restrict__ p, long long n4) {
    long long i = (long long)blockIdx.x * blockDim.x + threadIdx.x;
    if (i < n4) reinterpret_cast<float4*>(p)[i] = make_float4(0.f, 0.f, 0.f, 0.f);
}

// ------------------------------------------- KG edge scatter: wave per edge
__global__ void edge_scatter(const int* __restrict__ edge_index,
                             const int* __restrict__ edge_type,
                             const float* __restrict__ ent,
                             const float* __restrict__ weight,
                             float* __restrict__ agg,
                             float* __restrict__ cnt) {
    long long tid = (long long)blockIdx.x * blockDim.x + threadIdx.x;
    int e    = (int)(tid >> 5);
    int lane = (int)(tid & 31);
    if (e >= N_EDGES) return;
    int head = edge_index[e];
    int tail = edge_index[N_EDGES + e];
    int rel  = edge_type[e];
    int d = lane * 2;
    float2 w = *reinterpret_cast<const float2*>(weight + rel * DIM + d);
    float2 x = *reinterpret_cast<const float2*>(ent + (size_t)tail * DIM + d);
    atomicAdd(&agg[(size_t)head * DIM + d],     x.x * w.x);
    atomicAdd(&agg[(size_t)head * DIM + d + 1], x.y * w.y);
    if (lane == 0) atomicAdd(&cnt[head], 1.0f);
}

// --------------------------------- user->item scatter (weight row 0), wave/nnz
__global__ void ui_scatter(const int* __restrict__ mat_row,
                           const int* __restrict__ mat_col,
                           const float* __restrict__ user,
                           const float* __restrict__ weight,
                           float* __restrict__ agg,
                           float* __restrict__ cnt) {
    long long tid = (long long)blockIdx.x * blockDim.x + threadIdx.x;
    int n    = (int)(tid >> 5);
    int lane = (int)(tid & 31);
    if (n >= NNZ) return;
    int r = mat_row[n];
    int c = mat_col[n];
    int d = lane * 2;
    float2 w = *reinterpret_cast<const float2*>(weight + d);  // relation 0
    float2 x = *reinterpret_cast<const float2*>(user + (size_t)r * DIM + d);
    atomicAdd(&agg[(size_t)c * DIM + d],     x.x * w.x);
    atomicAdd(&agg[(size_t)c * DIM + d + 1], x.y * w.y);
    if (lane == 0) atomicAdd(&cnt[c], 1.0f);
}

// ------------------------- gate GEMM + sigmoid fusion via V_WMMA_F32_16X16X4_F32
// gi = sigmoid(item_kg @ W1^T + i_u @ W2^T); fusion = gi*item_kg + (1-gi)*i_u
// One wave computes a 16-row tile.  fusion may alias iu_agg (rows staged first).
#define GWAVES 2
__global__ __launch_bounds__(GWAVES * 32)
void gate_fusion_kernel(const float* __restrict__ entity_agg,
                        const float* __restrict__ ent_cnt,
                        const float* __restrict__ iu_agg,
                        const float* __restrict__ item_cnt,
                        const float* __restrict__ W1,
                        const float* __restrict__ W2,
                        float* __restrict__ fusion) {
    __shared__ float Ws1[DIM][68];           // W[n][k], padded stride (bank-safe)
    __shared__ float Ws2[DIM][68];
    __shared__ float As1[GWAVES][16][68];    // normalized item_kg tile
    __shared__ float As2[GWAVES][16][68];    // normalized i_u tile

    const int t = threadIdx.x;
    for (int i = t; i < DIM * DIM; i += GWAVES * 32) {
        Ws1[i >> 6][i & 63] = W1[i];
        Ws2[i >> 6][i & 63] = W2[i];
    }

    const int wave = t >> 5;
    const int lane = t & 31;
    const int NT   = N_ITEMS / 16;                    // 3125, exact
    const int tile = blockIdx.x * GWAVES + wave;
    const bool active = tile < NT;
    const int tilec = active ? tile : (NT - 1);       // keep all waves uniform
    const int r0 = tilec * 16;

    // stage + normalize the 16x64 A tiles (scatter-mean divide happens here)
    for (int i = lane; i < 16 * DIM; i += 32) {
        int m = i >> 6, k = i & 63;
        int row = r0 + m;
        float ck = fmaxf(ent_cnt[row], 1.0f);
        float cu = fmaxf(item_cnt[row], 1.0f);
        As1[wave][m][k] = entity_agg[(size_t)row * DIM + k] / ck;
        As2[wave][m][k] = iu_agg[(size_t)row * DIM + k] / cu;
    }
    __syncthreads();   // uniform: no wave returned early

    const int m    = lane & 15;       // A row
    const int half = lane >> 4;       // 0/1 -> K offset of 2 (A/B), M offset of 8 (C)
    const int kh   = half * 2;

    for (int n0 = 0; n0 < DIM; n0 += 16) {
        const int n = n0 + (lane & 15);   // B row (output column)
        v8f c = {0.f, 0.f, 0.f, 0.f, 0.f, 0.f, 0.f, 0.f};
#pragma unroll
        for (int kk = 0; kk < 16; ++kk) {
            const int kb = kk * 4 + kh;
            v2f a1 = { As1[wave][m][kb], As1[wave][m][kb + 1] };
            v2f b1 = { Ws1[n][kb],       Ws1[n][kb + 1] };
            c = __builtin_amdgcn_wmma_f32_16x16x4_f32(false, a1, false, b1,
                                                      (short)0, c, false, false);
            v2f a2 = { As2[wave][m][kb], As2[wave][m][kb + 1] };
            v2f b2 = { Ws2[n][kb],       Ws2[n][kb + 1] };
            c = __builtin_amdgcn_wmma_f32_16x16x4_f32(false, a2, false, b2,
                                                      (short)0, c, false, false);
        }
        if (active) {
#pragma unroll
            for (int v = 0; v < 8; ++v) {
                int mm = v + half * 8;                 // C layout row
                float z = c[v];
                float g = 1.0f / (1.0f + __expf(-z));
                float f = g * As1[wave][mm][n] + (1.0f - g) * As2[wave][mm][n];
                fusion[(size_t)(r0 + mm) * DIM + n] = f;
            }
        }
    }
}

// ----------------------------- user_agg = segment_sum(fusion[mat_col], mat_row)
__global__ void user_scatter(const int* __restrict__ mat_row,
                             const int* __restrict__ mat_col,
                             const float* __restrict__ fusion,
                             float* __restrict__ uagg) {
    long long tid = (long long)blockIdx.x * blockDim.x + threadIdx.x;
    int n    = (int)(tid >> 5);
    int lane = (int)(tid & 31);
    if (n >= NNZ) return;
    int r = mat_row[n];
    int c = mat_col[n];
    int d = lane * 2;
    float2 f = *reinterpret_cast<const float2*>(fusion + (size_t)c * DIM + d);
    atomicAdd(&uagg[(size_t)r * DIM + d],     f.x);
    atomicAdd(&uagg[(size_t)r * DIM + d + 1], f.y);
}

// --------------------- entity_emb = l2norm(concat(fusion, att_agg/cnt)); res +=
__global__ void finalize_entity(const float* __restrict__ fusion,
                                const float* __restrict__ agg,
                                const float* __restrict__ cnt,
                                float* __restrict__ ent_cur,
                                float* __restrict__ ent_res) {
    long long tid = (long long)blockIdx.x * blockDim.x + threadIdx.x;
    int row  = (int)(tid >> 5);
    int lane = (int)(tid & 31);
    if (row >= N_ENTITIES) return;
    int d = lane * 2;
    float v0, v1;
    if (row < N_ITEMS) {
        float2 f = *reinterpret_cast<const float2*>(fusion + (size_t)row * DIM + d);
        v0 = f.x; v1 = f.y;
    } else {
        float c = fmaxf(cnt[row], 1.0f);
        float2 a = *reinterpret_cast<const float2*>(agg + (size_t)row * DIM + d);
        v0 = a.x / c; v1 = a.y / c;
    }
    float s = v0 * v0 + v1 * v1;
#pragma unroll
    for (int off = 16; off > 0; off >>= 1) s += __shfl_xor(s, off, 32);
    float inv = 1.0f / fmaxf(sqrtf(s), 1e-12f);
    v0 *= inv; v1 *= inv;
    *reinterpret_cast<float2*>(ent_cur + (size_t)row * DIM + d) = make_float2(v0, v1);
    ent_res[(size_t)row * DIM + d]     += v0;
    ent_res[(size_t)row * DIM + d + 1] += v1;
}

__global__ void finalize_user(const float* __restrict__ uagg,
                              float* __restrict__ ucur,
                              float* __restrict__ user_res) {
    long long tid = (long long)blockIdx.x * blockDim.x + threadIdx.x;
    int row  = (int)(tid >> 5);
    int lane = (int)(tid & 31);
    if (row >= N_USERS) return;
    int d = lane * 2;
    float2 a = *reinterpret_cast<const float2*>(uagg + (size_t)row * DIM + d);
    float v0 = a.x, v1 = a.y;
    float s = v0 * v0 + v1 * v1;
#pragma unroll
    for (int off = 16; off > 0; off >>= 1) s += __shfl_xor(s, off, 32);
    float inv = 1.0f / fmaxf(sqrtf(s), 1e-12f);
    v0 *= inv; v1 *= inv;
    *reinterpret_cast<float2*>(ucur + (size_t)row * DIM + d) = make_float2(v0, v1);
    user_res[(size_t)row * DIM + d]     += v0;
    user_res[(size_t)row * DIM + d + 1] += v1;
}

// -------------------------------------------------------------------- launcher
extern "C" void kernel_launch(void* const* d_in, const int* in_sizes, int n_in,
                              void* d_out, int out_size, void* d_ws, size_t ws_size,
                              hipStream_t stream) {
    const float* user_in  = (const float*)d_in[0];
    const float* ent_in   = (const float*)d_in[1];
    const float* weight   = (const float*)d_in[2];
    const float* gate1_w  = (const float*)d_in[3];
    const float* gate2_w  = (const float*)d_in[4];
    const int*   edge_idx = (const int*)d_in[5];
    const int*   edge_typ = (const int*)d_in[6];
    const int*   mat_row  = (const int*)d_in[7];
    const int*   mat_col  = (const int*)d_in[8];

    float* ent_res  = (float*)d_out;                               // 200000*64
    float* user_res = ent_res + (size_t)N_ENTITIES * DIM;          // 100000*64

    float* ws = (float*)d_ws;
    size_t o = 0;
    float* ent_cur    = ws + o; o += (size_t)N_ENTITIES * DIM;
    float* ucur       = ws + o; o += (size_t)N_USERS * DIM;
    float* entity_agg = ws + o; o += (size_t)N_ENTITIES * DIM;     // zero region start
    float* ent_cnt    = ws + o; o += (size_t)N_ENTITIES;
    float* iu_agg     = ws + o; o += (size_t)N_ITEMS * DIM;        // doubles as fusion
    float* item_cnt   = ws + o; o += (size_t)N_ITEMS;
    float* uagg       = ws + o; o += (size_t)N_USERS * DIM;
    const size_t zero_floats = (size_t)N_ENTITIES * DIM + N_ENTITIES
                             + (size_t)N_ITEMS * DIM + N_ITEMS
                             + (size_t)N_USERS * DIM;              // contiguous

    hipMemcpyAsync(ent_cur,  ent_in,  sizeof(float) * (size_t)N_ENTITIES * DIM,
                   hipMemcpyDeviceToDevice, stream);
    hipMemcpyAsync(ucur,     user_in, sizeof(float) * (size_t)N_USERS * DIM,
                   hipMemcpyDeviceToDevice, stream);
    hipMemcpyAsync(ent_res,  ent_in,  sizeof(float) * (size_t)N_ENTITIES * DIM,
                   hipMemcpyDeviceToDevice, stream);
    hipMemcpyAsync(user_res, user_in, sizeof(float) * (size_t)N_USERS * DIM,
                   hipMemcpyDeviceToDevice, stream);

    for (int hop = 0; hop < N_HOPS; ++hop) {
        long long n4 = (long long)(zero_floats / 4);
        zero_kernel<<<(int)((n4 + 255) / 256), 256, 0, stream>>>(entity_agg, n4);

        {   long long th = (long long)N_EDGES * 32;
            edge_scatter<<<(int)((th + 255) / 256), 256, 0, stream>>>(
                edge_idx, edge_typ, ent_cur, weight, entity_agg, ent_cnt); }

        {   long long th = (long long)NNZ * 32;
            ui_scatter<<<(int)((th + 255) / 256), 256, 0, stream>>>(
                mat_row, mat_col, ucur, weight, iu_agg, item_cnt); }

        {   int blocks = (N_ITEMS / 16 + GWAVES - 1) / GWAVES;
            gate_fusion_kernel<<<blocks, GWAVES * 32, 0, stream>>>(
                entity_agg, ent_cnt, iu_agg, item_cnt,
                gate1_w + (size_t)hop * DIM * DIM,
                gate2_w + (size_t)hop * DIM * DIM,
                iu_agg /* fusion in place */); }

        {   long long th = (long long)NNZ * 32;
            user_scatter<<<(int)((th + 255) / 256), 256, 0, stream>>>(
                mat_row, mat_col, iu_agg, uagg); }

        {   long long th = (long long)N_ENTITIES * 32;
            finalize_entity<<<(int)((th + 255) / 256), 256, 0, stream>>>(
                iu_agg, entity_agg, ent_cnt, ent_cur, ent_res); }

        {   long long th = (long long)N_USERS * 32;
            finalize_user<<<(int)((th + 255) / 256), 256, 0, stream>>>(
                uagg, ucur, user_res); }
    }
}